// DWT_2D_16449724745580
// MI455X (gfx1250) — compile-verified
//
#include <hip/hip_runtime.h>
#include <stdint.h>

// ---------------------------------------------------------------------------
// 2D Haar DWT, (8,96,512,512) f32 -> 4 x (8,96,256,256) f32 (LL,LH,HL,HH).
// Memory-bound (~0.37 flop/byte): floor = 1.61 GB / 23.3 TB/s ~= 69 us.
// Data movement via the CDNA5 Tensor Data Mover (tensor_load_to_lds),
// double-buffered in LDS, pipelined with TENSORcnt.
// ---------------------------------------------------------------------------

typedef __attribute__((ext_vector_type(4))) unsigned int v4u;
typedef __attribute__((ext_vector_type(8))) int          v8i;
typedef __attribute__((ext_vector_type(4))) int          v4i;
typedef __attribute__((ext_vector_type(2))) float        v2f;
typedef __attribute__((ext_vector_type(4))) float        v4f;

#define WG_THREADS   256
#define TILE_ROWS    16
#define IMG_W        512
#define IMG_H        512
#define PLANES       768                       // 8 * 96
#define ROWS_PER_WG  128
#define TILES_PER_WG (ROWS_PER_WG / TILE_ROWS) // 8
#define TILE_ELEMS   (TILE_ROWS * IMG_W)       // 8192 floats = 32 KB
#define OUT_W        256
#define OUT_H        256

// Issue one TDM descriptor: contiguous 1D copy of TILE_ELEMS f32 from global
// into LDS at byte offset lds_byte_off. D# field layout per CDNA5 ISA §8.3/8.4.
// This toolchain's builtin takes 6 args: (g0 u32x4, g1 i32x8, g2 i32x4,
// g3 i32x4, extra i32x8, cpol i32) — unused groups are zero for a 1D copy.
__device__ __forceinline__ void tdm_load_tile(const float* gsrc, unsigned lds_byte_off) {
  const uint64_t ga = (uint64_t)(uintptr_t)gsrc;
  v4u g0;
  g0.x = 1u;                                                   // count=1 (valid user D#)
  g0.y = lds_byte_off;                                         // lds_addr   [63:32]
  g0.z = (unsigned)ga;                                         // global_addr[95:64]
  g0.w = (unsigned)((ga >> 32) & 0x01FFFFFFull) | (2u << 30);  // ga[56:32] | type=2
  v8i g1;
  g1[0] = (int)(2u << 16);                                     // data_size=2 (4B), wg_mask=0
  g1[1] = (int)(((unsigned)TILE_ELEMS & 0xFFFFu) << 16);       // tensor_dim0[15:0] @ [63:48]
  g1[2] = (int)((((unsigned)TILE_ELEMS >> 16) & 0xFFFFu)       // tensor_dim0[31:16]
                | (1u << 16));                                 // tensor_dim1 = 1
  g1[3] = (int)(((unsigned)TILE_ELEMS & 0xFFFFu) << 16);       // tile_dim0 = 8192 @ [127:112]
  g1[4] = 1;                                                   // tile_dim1=1, tile_dim2=0
  g1[5] = (int)TILE_ELEMS;                                     // tensor_dim0_stride[31:0]
  g1[6] = 0;
  g1[7] = 0;
  v4i gz4 = {0, 0, 0, 0};                                      // groups 2/3 unused (<=2D)
  v8i gz8 = {0, 0, 0, 0, 0, 0, 0, 0};                          // extra group (unused)
  __builtin_amdgcn_tensor_load_to_lds(g0, g1, gz4, gz4, gz8, 0);
}

__global__ __launch_bounds__(WG_THREADS)
void dwt_haar_tdm_kernel(const float* __restrict__ x, float* __restrict__ out) {
  __shared__ __align__(128) float smem[2][TILE_ELEMS];         // 64 KB double buffer

  const int seg     = (int)blockIdx.x;                         // 0 .. 3071
  const int plane   = seg >> 2;                                // 0 .. 767
  const int rowbase = (seg & 3) * ROWS_PER_WG;                 // 0,128,256,384
  const float* planeBase = x + (size_t)plane * (IMG_W * IMG_H);

  // Flat shared pointer: low 32 bits are the LDS byte offset (ISA aperture rule).
  const unsigned lds0 = (unsigned)(uintptr_t)(&smem[0][0]);
  const unsigned lds1 = (unsigned)(uintptr_t)(&smem[1][0]);

  const int  tid    = (int)threadIdx.x;
  const bool issuer = (tid < 32);                              // wave 0 drives the TDM

  if (issuer) tdm_load_tile(planeBase + (size_t)rowbase * IMG_W, lds0);

  const size_t SUB = (size_t)PLANES * OUT_W * OUT_H;
  float* __restrict__ oLL = out;
  float* __restrict__ oLH = out + SUB;
  float* __restrict__ oHL = out + 2 * SUB;
  float* __restrict__ oHH = out + 3 * SUB;

  for (int t = 0; t < TILES_PER_WG; ++t) {
    if (issuer) {
      if (t + 1 < TILES_PER_WG) {
        // Prefetch next tile into the other buffer, then ensure tile t landed.
        tdm_load_tile(planeBase + (size_t)(rowbase + (t + 1) * TILE_ROWS) * IMG_W,
                      (t & 1) ? lds0 : lds1);
        __builtin_amdgcn_s_wait_tensorcnt(1);   // in-order: <=1 outstanding => tile t done
      } else {
        __builtin_amdgcn_s_wait_tensorcnt(0);
      }
    }
    __syncthreads();                            // publish tile t to all waves

    const float* buf = (t & 1) ? &smem[1][0] : &smem[0][0];

    // Tile = 8 row-pairs x 128 column-quads = 1024 units; 4 units per thread.
    // Each unit: two ds_load_b128 (even/odd row) -> two 2x2 butterflies.
#pragma unroll
    for (int k = 0; k < (TILE_ROWS / 2) * (IMG_W / 4) / WG_THREADS; ++k) {
      const int u  = tid + k * WG_THREADS;
      const int pr = u >> 7;                    // row pair 0..7
      const int cu = u & 127;                   // column quad 0..127
      const v4f r0 = *(const v4f*)(buf + (2 * pr)     * IMG_W + 4 * cu);
      const v4f r1 = *(const v4f*)(buf + (2 * pr + 1) * IMG_W + 4 * cu);

      // Horizontal butterflies (in-lane, no shuffles needed).
      const float s0 = r0.x + r0.y, d0 = r0.x - r0.y;   // even row, pair 0
      const float s1 = r0.z + r0.w, d1 = r0.z - r0.w;   // even row, pair 1
      const float t0 = r1.x + r1.y, e0 = r1.x - r1.y;   // odd  row, pair 0
      const float t1 = r1.z + r1.w, e1 = r1.z - r1.w;   // odd  row, pair 1

      // Vertical butterflies + 1/2 scaling (outer products of [s,s]/[s,-s]).
      const v2f ll = {0.5f * (s0 + t0), 0.5f * (s1 + t1)};
      const v2f lh = {0.5f * (s0 - t0), 0.5f * (s1 - t1)};
      const v2f hl = {0.5f * (d0 + e0), 0.5f * (d1 + e1)};
      const v2f hh = {0.5f * (d0 - e0), 0.5f * (d1 - e1)};

      const int    orow = (rowbase >> 1) + t * (TILE_ROWS / 2) + pr;
      const size_t ob   = (((size_t)plane * OUT_H) + orow) * OUT_W + 2 * cu;

      // Streamed once: non-temporal stores, coalesced 8B per lane.
      __builtin_nontemporal_store(ll, (v2f*)(oLL + ob));
      __builtin_nontemporal_store(lh, (v2f*)(oLH + ob));
      __builtin_nontemporal_store(hl, (v2f*)(oHL + ob));
      __builtin_nontemporal_store(hh, (v2f*)(oHH + ob));
    }
    __syncthreads();                            // protect buf[t&1] before it is re-filled
  }
}

extern "C" void kernel_launch(void* const* d_in, const int* in_sizes, int n_in,
                              void* d_out, int out_size, void* d_ws, size_t ws_size,
                              hipStream_t stream) {
  (void)in_sizes; (void)n_in; (void)d_ws; (void)ws_size; (void)out_size;
  const float* x  = (const float*)d_in[0];
  float* out      = (float*)d_out;
  dim3 grid(PLANES * (IMG_H / ROWS_PER_WG));    // 768 * 4 = 3072 workgroups
  dim3 block(WG_THREADS);
  hipLaunchKernelGGL(dwt_haar_tdm_kernel, grid, block, 0, stream, x, out);
}